// TWNQuantizer_37134287241931
// MI455X (gfx1250) — compile-verified
//
#include <hip/hip_runtime.h>
#include <stdint.h>

typedef float v4f __attribute__((ext_vector_type(4)));
typedef int   v4i __attribute__((ext_vector_type(4)));

#define AS1 __attribute__((address_space(1)))
#define AS3 __attribute__((address_space(3)))

// ---------------- CDNA5 async global->LDS copy (128-bit per lane) ----------
__device__ __forceinline__ void async_copy_g2l_b128(void* g, void* l) {
#if __has_builtin(__builtin_amdgcn_global_load_async_to_lds_b128)
  __builtin_amdgcn_global_load_async_to_lds_b128((AS1 v4i*)g, (AS3 v4i*)l, 0, 0);
#else
  unsigned int loff = (unsigned int)(uintptr_t)(AS3 void*)l;
  asm volatile("global_load_async_to_lds_b128 %0, %1, off"
               :: "v"(loff), "v"(g) : "memory");
#endif
}

__device__ __forceinline__ void wait_asynccnt0() {
#if __has_builtin(__builtin_amdgcn_s_wait_asynccnt)
  __builtin_amdgcn_s_wait_asynccnt(0);
#else
  asm volatile("s_wait_asynccnt 0" ::: "memory");
#endif
}

// ---------------- deterministic block sum (wave32) --------------------------
__device__ __forceinline__ float block_sum_to_t0(float v, float* wred) {
#pragma unroll
  for (int off = 16; off > 0; off >>= 1) v += __shfl_xor(v, off, 32);
  const int t = threadIdx.x;
  if ((t & 31) == 0) wred[t >> 5] = v;
  __syncthreads();
  float s = 0.0f;
  if (t == 0) {
    const int nw = blockDim.x >> 5;
    for (int j = 0; j < nw; ++j) s += wred[j];
  }
  __syncthreads();  // make wred reusable by a second reduction
  return s;
}

// ---------------- Kernel X: per-row int8 quant-dequant ----------------------
// One block per 4096-float row. Row staged into LDS with async b128 copies,
// consumed twice (abs-max reduce, then dequant). NT store: pure stream.
__global__ __launch_bounds__(256) void quant_x_kernel(const float* __restrict__ x,
                                                      float* __restrict__ out) {
  __shared__ v4f  smem[1024];   // 16 KB row tile
  __shared__ float wred[8];
  const int t = threadIdx.x;
  const long long row = blockIdx.x;
  const v4f* __restrict__ src4 = (const v4f*)x + row * 1024;
  v4f* __restrict__ dst4 = (v4f*)out + row * 1024;

#pragma unroll
  for (int i = 0; i < 4; ++i) {
    const int v = i * 256 + t;
    async_copy_g2l_b128((void*)(src4 + v), (void*)&smem[v]);
  }
  wait_asynccnt0();
  __syncthreads();

  v4f vals[4];
  float m = 0.0f;
#pragma unroll
  for (int i = 0; i < 4; ++i) {
    vals[i] = smem[i * 256 + t];  // ds_load_b128
    const v4f a = vals[i];
    m = fmaxf(m, fmaxf(fmaxf(fabsf(a.x), fabsf(a.y)),
                       fmaxf(fabsf(a.z), fabsf(a.w))));
  }
#pragma unroll
  for (int off = 16; off > 0; off >>= 1) m = fmaxf(m, __shfl_xor(m, off, 32));
  if ((t & 31) == 0) wred[t >> 5] = m;
  __syncthreads();
  m = wred[0];
#pragma unroll
  for (int j = 1; j < 8; ++j) m = fmaxf(m, wred[j]);

  const float maxv  = fmaxf(m, 1e-5f);
  const float scale = maxv * (1.0f / 127.0f);
  const float inv   = 127.0f / maxv;

#pragma unroll
  for (int i = 0; i < 4; ++i) {
    const v4f a = vals[i];
    v4f q;
    q.x = fminf(fmaxf(rintf(a.x * inv), -128.0f), 127.0f) * scale;
    q.y = fminf(fmaxf(rintf(a.y * inv), -128.0f), 127.0f) * scale;
    q.z = fminf(fmaxf(rintf(a.z * inv), -128.0f), 127.0f) * scale;
    q.w = fminf(fmaxf(rintf(a.w * inv), -128.0f), 127.0f) * scale;
    __builtin_nontemporal_store(q, dst4 + i * 256 + t);
  }
}

// ---------------- W pass 1: per-block partial sum of |w| --------------------
__global__ __launch_bounds__(256) void w_abssum_partial(const float* __restrict__ wp,
                                                        float* __restrict__ partial,
                                                        int n4) {
  __shared__ float wred[8];
  const v4f* w4 = (const v4f*)wp;
  float s = 0.0f;
  for (int i = blockIdx.x * blockDim.x + threadIdx.x; i < n4;
       i += gridDim.x * blockDim.x) {
    const v4f a = w4[i];
    s += fabsf(a.x) + fabsf(a.y) + fabsf(a.z) + fabsf(a.w);
  }
  const float tot = block_sum_to_t0(s, wred);
  if (threadIdx.x == 0) partial[blockIdx.x] = tot;
}

// ---------------- W pass 2: reduce partials -> delta ------------------------
__global__ __launch_bounds__(256) void w_delta_kernel(const float* __restrict__ partial,
                                                      int nblk, float n_elems,
                                                      float* __restrict__ ws) {
  __shared__ float wred[8];
  float s = 0.0f;
  for (int i = threadIdx.x; i < nblk; i += blockDim.x) s += partial[i];
  const float tot = block_sum_to_t0(s, wred);
  if (threadIdx.x == 0) {
    const float mean = tot / n_elems;
    ws[0] = fmaxf(0.75f * mean, 1e-5f);  // delta
  }
}

// ---------------- W pass 3: masked partial sum + count ----------------------
__global__ __launch_bounds__(256) void w_masked_partial(const float* __restrict__ wp,
                                                        const float* __restrict__ ws,
                                                        float* __restrict__ psum,
                                                        float* __restrict__ pcnt,
                                                        int n4) {
  __shared__ float wred[8];
  const float delta = ws[0];
  const v4f* w4 = (const v4f*)wp;
  float s = 0.0f, c = 0.0f;
  for (int i = blockIdx.x * blockDim.x + threadIdx.x; i < n4;
       i += gridDim.x * blockDim.x) {
    const v4f a = w4[i];
    float ax;
    ax = fabsf(a.x); if (ax > delta) { s += ax; c += 1.0f; }
    ax = fabsf(a.y); if (ax > delta) { s += ax; c += 1.0f; }
    ax = fabsf(a.z); if (ax > delta) { s += ax; c += 1.0f; }
    ax = fabsf(a.w); if (ax > delta) { s += ax; c += 1.0f; }
  }
  const float ts = block_sum_to_t0(s, wred);
  const float tc = block_sum_to_t0(c, wred);
  if (threadIdx.x == 0) { psum[blockIdx.x] = ts; pcnt[blockIdx.x] = tc; }
}

// ---------------- W pass 4: reduce -> scale ---------------------------------
__global__ __launch_bounds__(256) void w_scale_kernel(const float* __restrict__ psum,
                                                      const float* __restrict__ pcnt,
                                                      int nblk,
                                                      float* __restrict__ ws) {
  __shared__ float wred[8];
  float s = 0.0f, c = 0.0f;
  for (int i = threadIdx.x; i < nblk; i += blockDim.x) { s += psum[i]; c += pcnt[i]; }
  const float ts = block_sum_to_t0(s, wred);
  const float tc = block_sum_to_t0(c, wred);
  if (threadIdx.x == 0)
    ws[1] = (tc > 0.0f) ? (ts / fmaxf(tc, 1.0f)) : 0.0f;  // scale
}

// ---------------- W pass 5: ternarize + dequant -----------------------------
__global__ __launch_bounds__(256) void w_ternarize(const float* __restrict__ wp,
                                                   const float* __restrict__ ws,
                                                   float* __restrict__ outp,
                                                   int n4) {
  const float delta = ws[0];
  const float scale = ws[1];
  const int i = blockIdx.x * blockDim.x + threadIdx.x;
  if (i >= n4) return;
  const v4f a = ((const v4f*)wp)[i];
  v4f q;
  q.x = (a.x > delta) ? scale : ((a.x < -delta) ? -scale : 0.0f);
  q.y = (a.y > delta) ? scale : ((a.y < -delta) ? -scale : 0.0f);
  q.z = (a.z > delta) ? scale : ((a.z < -delta) ? -scale : 0.0f);
  q.w = (a.w > delta) ? scale : ((a.w < -delta) ? -scale : 0.0f);
  __builtin_nontemporal_store(q, (v4f*)outp + i);
}

// ---------------- launch ----------------------------------------------------
extern "C" void kernel_launch(void* const* d_in, const int* in_sizes, int n_in,
                              void* d_out, int out_size, void* d_ws, size_t ws_size,
                              hipStream_t stream) {
  const float* x = (const float*)d_in[0];   // (4,4096,4096) f32
  const float* w = (const float*)d_in[1];   // (4096,4096)  f32
  float* out = (float*)d_out;

  const int nx = in_sizes[0];               // 67108864
  const int nw = in_sizes[1];               // 16777216
  const int rows = nx / 4096;               // 16384 rows of 4096
  const int n4 = nw / 4;

  float* xout = out;
  float* wout = out + nx;

  float* ws = (float*)d_ws;                 // [0]=delta [1]=scale, then partials
  const int NBLK = 1024;
  float* pAbs = ws + 16;
  float* pSum = pAbs + NBLK;
  float* pCnt = pSum + NBLK;

  // x first (streams 512 MB with NT stores), then the w passes back-to-back so
  // w (64 MB) stays resident in the 192 MB L2 across its three reads.
  quant_x_kernel<<<rows, 256, 0, stream>>>(x, xout);
  w_abssum_partial<<<NBLK, 256, 0, stream>>>(w, pAbs, n4);
  w_delta_kernel<<<1, 256, 0, stream>>>(pAbs, NBLK, (float)nw, ws);
  w_masked_partial<<<NBLK, 256, 0, stream>>>(w, ws, pSum, pCnt, n4);
  w_scale_kernel<<<1, 256, 0, stream>>>(pSum, pCnt, NBLK, ws);
  w_ternarize<<<(n4 + 255) / 256, 256, 0, stream>>>(w, ws, wout, n4);
}